// Decoder_27049704031064
// MI455X (gfx1250) — compile-verified
//
#include <hip/hip_runtime.h>
#include <math.h>

#define B_ 8
#define T_ 512
#define D_ 1024
#define H_ 16
#define DH_ 64
#define L_ 4
#define C_ 10
#define EPS_ 1e-5f
#define INV_RD (1.0f/32.0f)   // 1/sqrt(1024)
#define GKC 32

typedef __attribute__((ext_vector_type(2))) float v2f;
typedef __attribute__((ext_vector_type(8))) float v8f;
typedef unsigned int u32x4 __attribute__((ext_vector_type(4)));
typedef int i32x4 __attribute__((ext_vector_type(4)));
typedef int i32x8 __attribute__((ext_vector_type(8)));
typedef int avi4 __attribute__((vector_size(4 * sizeof(int))));  // builtin's pointee type

#define AS1 __attribute__((address_space(1)))
#define AS3 __attribute__((address_space(3)))

// ---- CDNA5 data-mover feature detection (device pass only; host pass falls back)
#if defined(__has_builtin)
#  if __has_builtin(__builtin_amdgcn_tensor_load_to_lds) && \
      __has_builtin(__builtin_amdgcn_s_wait_tensorcnt)
#    define USE_TDM 1
#  endif
#  if __has_builtin(__builtin_amdgcn_global_load_async_to_lds_b128) && \
      __has_builtin(__builtin_amdgcn_s_wait_asynccnt)
#    define USE_ASYNC 1
#  endif
#endif
#ifndef USE_TDM
#  define USE_TDM 0
#endif
#ifndef USE_ASYNC
#  define USE_ASYNC 0
#endif

static __device__ __forceinline__ v8f wmma_f32(v2f a, v2f b, v8f c) {
  // D = A(16x4,f32) * B(4x16,f32) + C(16x16,f32)
  return __builtin_amdgcn_wmma_f32_16x16x4_f32(
      /*neg_a=*/false, a, /*neg_b=*/false, b,
      /*c_mod=*/(short)0, c, /*reuse_a=*/false, /*reuse_b=*/false);
}

#if USE_TDM
// TDM 2D tile load: 32x64 f32 tile, row stride 1024 f32, LDS rows padded
// 32dw + 4dw (matches __shared__ float [64][36]).  D# packing per
// CDNA5 ISA.
static __device__ __forceinline__ void tdm_load_tile(const float* gtile,
                                                     unsigned lds_off,
                                                     unsigned rows_total) {
  unsigned long long ga = (unsigned long long)(size_t)gtile;
  u32x4 g0;
  g0[0] = 1u;                                         // count=1, user mode
  g0[1] = lds_off;                                    // lds_addr (bytes)
  g0[2] = (unsigned)(ga & 0xFFFFFFFFu);               // global_addr[31:0]
  g0[3] = (unsigned)((ga >> 32) & 0x01FFFFFFu)        // global_addr[56:32]
          | 0x80000000u;                              // type=2 ("image")
  i32x8 g1;
  g1[0] = (int)((2u << 16)        // data_size = 4 bytes
              | (1u << 20)        // pad_enable
              | (4u << 22)        // pad_interval: 32 dwords
              | (3u << 25));      // pad_amount:   4 dwords
  g1[1] = (int)(1024u << 16);                         // tensor_dim0 = 1024 (K)
  g1[2] = (int)((rows_total & 0xFFFFu) << 16);        // tensor_dim1 lo16
  g1[3] = (int)((32u << 16) | (rows_total >> 16));    // tile_dim0=32 | dim1 hi16
  g1[4] = 64;                                         // tile_dim1 = 64 rows
  g1[5] = 1024;                                       // tensor_dim0_stride
  g1[6] = 0;
  g1[7] = 0;                                          // dim1_stride unused (2D)
  i32x4 z4 = {0, 0, 0, 0};
#if __clang_major__ >= 23
  i32x8 z8 = {0, 0, 0, 0, 0, 0, 0, 0};
  __builtin_amdgcn_tensor_load_to_lds(g0, g1, z4, z4, z8, 0);
#else
  __builtin_amdgcn_tensor_load_to_lds(g0, g1, z4, z4, 0);
#endif
}
#endif

#if USE_ASYNC
static __device__ __forceinline__ void async_cp_b128(const float* g, float* l) {
  // per-lane 16B global -> LDS, tracked by ASYNCcnt
  __builtin_amdgcn_global_load_async_to_lds_b128(
      (AS1 avi4*)(size_t)g,
      (AS3 avi4*)(unsigned)(size_t)l,
      0, 0);
}
#endif

// ---------------------------------------------------------------- pos add
__global__ void k_addpos(const float* __restrict__ xin,
                         const float* __restrict__ pos,
                         float* __restrict__ x) {
  size_t i = (size_t)blockIdx.x * 256 + threadIdx.x;
  x[i] = xin[i] + pos[i % ((size_t)T_ * D_)];
}

// ---------------------------------------------------------------- layernorm (in-place, biased var)
__global__ __launch_bounds__(256) void k_ln(float* __restrict__ x,
                                            const float* __restrict__ g,
                                            const float* __restrict__ b) {
  __shared__ float s1[256];
  __shared__ float s2[256];
  const int row = blockIdx.x;
  const int t = threadIdx.x;
  float4* xr = (float4*)(x + (size_t)row * D_);
  float4 v = xr[t];
  float sum = v.x + v.y + v.z + v.w;
  float sq  = v.x*v.x + v.y*v.y + v.z*v.z + v.w*v.w;
  s1[t] = sum; s2[t] = sq;
  __syncthreads();
  for (int off = 128; off > 0; off >>= 1) {
    if (t < off) { s1[t] += s1[t+off]; s2[t] += s2[t+off]; }
    __syncthreads();
  }
  const float mean = s1[0] * (1.0f / D_);
  const float var  = s2[0] * (1.0f / D_) - mean * mean;
  const float inv  = rsqrtf(var + EPS_);
  float4 gv = ((const float4*)g)[t];
  float4 bv = ((const float4*)b)[t];
  v.x = (v.x - mean) * inv * gv.x + bv.x;
  v.y = (v.y - mean) * inv * gv.y + bv.y;
  v.z = (v.z - mean) * inv * gv.z + bv.z;
  v.w = (v.w - mean) * inv * gv.w + bv.w;
  xr[t] = v;
}

// ---------------------------------------------------------------- GEMM  C[M,N] = A[M,K] * W[N,K]^T  (nn.Linear)
// mode 0: plain store   mode 1: relu(C + bias) + A  (fused FC epilogue)
__global__ __launch_bounds__(128) void k_gemm(const float* __restrict__ A,
                                              const float* __restrict__ W,
                                              const float* __restrict__ bias,
                                              float* __restrict__ Cmat,
                                              int mode, int mrows) {
  __shared__ float As[64][36];   // 36 = 32 + pad (= TDM pad 4dw/32dw)
  __shared__ float Ws[64][36];
  const int bm = blockIdx.y * 64, bn = blockIdx.x * 64;
  const int tid = threadIdx.x, lane = tid & 31, w = tid >> 5;
  const int hl = lane >> 4, l16 = lane & 15;
  const int wm = (w >> 1) * 32, wn = (w & 1) * 32;
  v8f acc[2][2] = {};
  for (int kc = 0; kc < D_; kc += GKC) {
#if USE_TDM
    if (tid < 32) {   // wave 0 drives the Tensor Data Mover
      tdm_load_tile(&A[(size_t)bm * D_ + kc],
                    (unsigned)(size_t)&As[0][0], (unsigned)mrows);
      tdm_load_tile(&W[(size_t)bn * D_ + kc],
                    (unsigned)(size_t)&Ws[0][0], 1024u);
      __builtin_amdgcn_s_wait_tensorcnt(0);
    }
#else
    #pragma unroll
    for (int p = 0; p < 4; ++p) {
      int j = tid + p * 128;
      int r = j >> 3, c4 = (j & 7) * 4;
      *(float4*)&As[r][c4] = *(const float4*)&A[(size_t)(bm + r) * D_ + kc + c4];
      *(float4*)&Ws[r][c4] = *(const float4*)&W[(size_t)(bn + r) * D_ + kc + c4];
    }
    if (kc + GKC < D_) {  // global_prefetch_b8 for next K-chunk
      __builtin_prefetch(&A[(size_t)(bm + (tid >> 1)) * D_ + kc + GKC], 0, 1);
      __builtin_prefetch(&W[(size_t)(bn + (tid >> 1)) * D_ + kc + GKC], 0, 1);
    }
#endif
    __syncthreads();
    #pragma unroll
    for (int kk = 0; kk < GKC; kk += 4) {
      v2f a0 = *(const v2f*)&As[wm +      l16][kk + 2*hl];
      v2f a1 = *(const v2f*)&As[wm + 16 + l16][kk + 2*hl];
      v2f b0 = *(const v2f*)&Ws[wn +      l16][kk + 2*hl];
      v2f b1 = *(const v2f*)&Ws[wn + 16 + l16][kk + 2*hl];
      acc[0][0] = wmma_f32(a0, b0, acc[0][0]);
      acc[0][1] = wmma_f32(a0, b1, acc[0][1]);
      acc[1][0] = wmma_f32(a1, b0, acc[1][0]);
      acc[1][1] = wmma_f32(a1, b1, acc[1][1]);
    }
    __syncthreads();
  }
  #pragma unroll
  for (int mi = 0; mi < 2; ++mi)
    #pragma unroll
    for (int ni = 0; ni < 2; ++ni)
      #pragma unroll
      for (int v = 0; v < 8; ++v) {
        int m = bm + wm + mi * 16 + v + 8 * hl;
        int n = bn + wn + ni * 16 + l16;
        float val = acc[mi][ni][v];
        if (mode == 1) {
          val += bias[n];
          val = fmaxf(val, 0.0f) + A[(size_t)m * D_ + n];
        }
        Cmat[(size_t)m * D_ + n] = val;
      }
}

// ---------------------------------------------------------------- fused flash attention (+residual into X)
// kv_bhtd=0: K/V laid out [B,T,D] (col = h*64+e).  kv_bhtd=1: [B,H,T,DH].
__global__ __launch_bounds__(128) void k_flash(const float* __restrict__ Q,
                                               const float* __restrict__ Kp,
                                               const float* __restrict__ Vp,
                                               float* __restrict__ X,
                                               int kv_bhtd, int causal) {
  __shared__ float Ks[16][68];
  __shared__ float Vs[16][68];
  __shared__ float Ps[4][16][17];
  const int tid = threadIdx.x, lane = tid & 31, w = tid >> 5;
  const int hl = lane >> 4, l16 = lane & 15;
  const int b = blockIdx.z, h = blockIdx.y, q0 = blockIdx.x * 64;
  const int qrow_base = q0 + w * 16;

  // resident Q fragments: A-matrix lane m = l%16, K pairs at 4*s + 2*hl
  v2f qf[16];
  const float* qp = Q + ((size_t)(b * T_ + qrow_base + l16)) * D_ + h * DH_;
  #pragma unroll
  for (int s = 0; s < 16; ++s) qf[s] = *(const v2f*)&qp[4 * s + 2 * hl];

  float rm[8], rs[8];
  #pragma unroll
  for (int v = 0; v < 8; ++v) { rm[v] = -3.0e38f; rs[v] = 0.0f; }
  v8f o[4] = {};

  const int nkc = causal ? (q0 / 16 + 4) : (T_ / 16);
  for (int kc = 0; kc < nkc; ++kc) {
    { // cooperative K/V tile load: 16x64 each
      int r = tid >> 3, c = (tid & 7) * 8;
      size_t base;
      if (kv_bhtd) base = ((size_t)((b * H_ + h) * T_) + kc * 16 + r) * DH_ + c;
      else         base = ((size_t)(b * T_ + kc * 16 + r)) * D_ + h * DH_ + c;
#if USE_ASYNC
      async_cp_b128(&Kp[base],     &Ks[r][c]);
      async_cp_b128(&Kp[base + 4], &Ks[r][c + 4]);
      async_cp_b128(&Vp[base],     &Vs[r][c]);
      async_cp_b128(&Vp[base + 4], &Vs[r][c + 4]);
      __builtin_amdgcn_s_wait_asynccnt(0);
#else
      *(float4*)&Ks[r][c]     = *(const float4*)&Kp[base];
      *(float4*)&Ks[r][c + 4] = *(const float4*)&Kp[base + 4];
      *(float4*)&Vs[r][c]     = *(const float4*)&Vp[base];
      *(float4*)&Vs[r][c + 4] = *(const float4*)&Vp[base + 4];
#endif
    }
    __syncthreads();

    // S = Q * K^T   (16x16 tile per wave, K-dim 64 -> 16 WMMA)
    v8f s = {};
    #pragma unroll
    for (int s16 = 0; s16 < 16; ++s16) {
      v2f bf = *(const v2f*)&Ks[l16][4 * s16 + 2 * hl];
      s = wmma_f32(qf[s16], bf, s);
    }

    // online softmax; rows m = v + 8*hl, cols kj = l16
    float p[8], f[8];
    #pragma unroll
    for (int v = 0; v < 8; ++v) {
      float sv = s[v] * INV_RD;
      if (causal) {
        int kj = kc * 16 + l16;
        int m  = qrow_base + v + 8 * hl;
        if (kj > m) sv = -3.0e38f;
      }
      float mx = sv;
      mx = fmaxf(mx, __shfl_xor(mx, 1, 32));
      mx = fmaxf(mx, __shfl_xor(mx, 2, 32));
      mx = fmaxf(mx, __shfl_xor(mx, 4, 32));
      mx = fmaxf(mx, __shfl_xor(mx, 8, 32));
      float nm = fmaxf(rm[v], mx);
      float pv = __expf(sv - nm);
      float ps = pv;
      ps += __shfl_xor(ps, 1, 32);
      ps += __shfl_xor(ps, 2, 32);
      ps += __shfl_xor(ps, 4, 32);
      ps += __shfl_xor(ps, 8, 32);
      f[v]  = __expf(rm[v] - nm);
      rs[v] = rs[v] * f[v] + ps;
      rm[v] = nm;
      p[v]  = pv;
    }
    #pragma unroll
    for (int nc = 0; nc < 4; ++nc)
      #pragma unroll
      for (int v = 0; v < 8; ++v) o[nc][v] *= f[v];

    // transpose P through per-wave LDS (C-layout -> A-layout)
    #pragma unroll
    for (int v = 0; v < 8; ++v) Ps[w][v + 8 * hl][l16] = p[v];

    // O += P * V   (K-dim 16 -> 4 WMMA per 16-col chunk)
    #pragma unroll
    for (int nc = 0; nc < 4; ++nc) {
      #pragma unroll
      for (int ks = 0; ks < 4; ++ks) {
        v2f a = *(const v2f*)&Ps[w][l16][4 * ks + 2 * hl];
        v2f bfr;
        bfr.x = Vs[4 * ks + 2 * hl    ][nc * 16 + l16];
        bfr.y = Vs[4 * ks + 2 * hl + 1][nc * 16 + l16];
        o[nc] = wmma_f32(a, bfr, o[nc]);
      }
    }
    __syncthreads();
  }

  // epilogue: X += O / rowsum   (residual)
  #pragma unroll
  for (int nc = 0; nc < 4; ++nc)
    #pragma unroll
    for (int v = 0; v < 8; ++v) {
      int m = qrow_base + v + 8 * hl;
      int d = h * DH_ + nc * 16 + l16;
      size_t idx = ((size_t)(b * T_ + m)) * D_ + d;
      X[idx] += o[nc][v] / rs[v];
    }
}

// ---------------------------------------------------------------- ordinal sigmoid head
__global__ void k_head(const float* __restrict__ X,
                       const float* __restrict__ cutoff,
                       float* __restrict__ out) {
  size_t i = (size_t)blockIdx.x * 256 + threadIdx.x;  // over B*T*D
  float xv = X[i];
  float bj = cutoff[0];
  float prev = 1.0f / (1.0f + __expf(-(bj - xv)));
  out[i * C_] = prev;
  #pragma unroll
  for (int j = 1; j < C_ - 1; ++j) {
    float cj = cutoff[j];
    bj += cj * cj;
    float sig = 1.0f / (1.0f + __expf(-(bj - xv)));
    out[i * C_ + j] = sig - prev;
    prev = sig;
  }
  out[i * C_ + (C_ - 1)] = 1.0f - prev;
}

// ---------------------------------------------------------------- launch
extern "C" void kernel_launch(void* const* d_in, const int* in_sizes, int n_in,
                              void* d_out, int out_size, void* d_ws, size_t ws_size,
                              hipStream_t stream) {
  const float* xin    = (const float*)d_in[0];
  const float* kin    = (const float*)d_in[1];
  const float* vin    = (const float*)d_in[2];
  const float* pos    = (const float*)d_in[3];
  const float* Wq1    = (const float*)d_in[4];
  const float* Wk1    = (const float*)d_in[5];
  const float* Wv1    = (const float*)d_in[6];
  const float* Wq2    = (const float*)d_in[7];
  const float* Wfc    = (const float*)d_in[8];
  const float* bfc    = (const float*)d_in[9];
  const float* g1     = (const float*)d_in[10];
  const float* b1     = (const float*)d_in[11];
  const float* g2     = (const float*)d_in[12];
  const float* b2     = (const float*)d_in[13];
  const float* g3     = (const float*)d_in[14];
  const float* b3     = (const float*)d_in[15];
  const float* cutoff = (const float*)d_in[16];
  float* out = (float*)d_out;

  const size_t N = (size_t)B_ * T_ * D_;
  float* x  = (float*)d_ws;
  float* q  = x + N;
  float* k1 = q + N;
  float* v1 = k1 + N;

  dim3 ggrid(D_ / 64, (B_ * T_) / 64);
  dim3 fgrid(T_ / 64, H_, B_);
  const int MR = B_ * T_;  // GEMM M rows (tensor_dim1 for TDM OOB)

  k_addpos<<<(int)(N / 256), 256, 0, stream>>>(xin, pos, x);

  for (int i = 0; i < L_; ++i) {
    size_t wOff = (size_t)i * D_ * D_;
    if (i > 0)
      k_ln<<<B_ * T_, 256, 0, stream>>>(x, g1 + (size_t)(i - 1) * D_, b1 + (size_t)(i - 1) * D_);
    k_gemm<<<ggrid, 128, 0, stream>>>(x, Wq1 + wOff, nullptr, q,  0, MR);
    k_gemm<<<ggrid, 128, 0, stream>>>(x, Wk1 + wOff, nullptr, k1, 0, MR);
    k_gemm<<<ggrid, 128, 0, stream>>>(x, Wv1 + wOff, nullptr, v1, 0, MR);
    k_flash<<<fgrid, 128, 0, stream>>>(q, k1, v1, x, 0, 0);          // self-attn, no mask
    k_ln<<<B_ * T_, 256, 0, stream>>>(x, g2 + (size_t)i * D_, b2 + (size_t)i * D_);
    k_gemm<<<ggrid, 128, 0, stream>>>(x, Wq2 + wOff, nullptr, q, 0, MR);
    k_flash<<<fgrid, 128, 0, stream>>>(q, kin, vin, x, 1, 1);        // cross-attn, causal
    k_ln<<<B_ * T_, 256, 0, stream>>>(x, g3 + (size_t)i * D_, b3 + (size_t)i * D_);
    k_gemm<<<ggrid, 128, 0, stream>>>(x, Wfc + wOff, bfc + (size_t)i * D_, x, 1, MR);  // fused FC
  }

  k_head<<<(int)(N / 256), 256, 0, stream>>>(x, cutoff, out);
}